// TriplaneSynthesisNetwork_61753039781917
// MI455X (gfx1250) — compile-verified
//
#include <hip/hip_runtime.h>

// ---------------------------------------------------------------------------
// Mip-pyramid warp (trilinear mip sampling), MI455X / gfx1250.
//
// Bandwidth-bound: ~270 MB total HBM traffic -> ~12us at 23.3 TB/s.
// - Pyramid downsample: separable [1,3,3,1]/8 blur. Horizontal pass in VALU
//   from an LDS-staged window (staged via global_load_async_to_lds_b32,
//   ASYNCcnt-tracked); vertical pass expressed as a banded 16x36 x 36x16
//   matmul on V_WMMA_F32_16X16X4_F32 (9 K-chunks of 4).
// - Warp kernel: one thread owns one output pixel, computes the 8 tap
//   addresses/weights once, loops all 96 channels (coalesced stores,
//   global_prefetch_b8 on the plane-strided tap walk).
//
// Workspace layout (floats): pyr1 (N*C*128^2) | pyr2 (N*C*64^2) |
//   pyr3 (N*C*32^2) | params (N*H*W float4)  => ~37.2 MB total.
// ---------------------------------------------------------------------------

typedef float v8f __attribute__((ext_vector_type(8)));
typedef float v2f __attribute__((ext_vector_type(2)));

__device__ __forceinline__ int refl(int i, int n) {
    // jnp.pad(mode='reflect') with pad=1: index -1 -> 1, index n -> n-2.
    return (i < 0) ? -i : ((i >= n) ? (2 * n - 2 - i) : i);
}

__device__ __forceinline__ float band_coef(int dy) {
    // [1,3,3,1]/8 separable blur tap, 0 outside the band.
    if (dy < 0 || dy > 3) return 0.0f;
    return (dy == 1 || dy == 2) ? 0.375f : 0.125f;
}

// One wave (32 lanes) produces one 16x16 output tile of the 2x blur-downsample.
__global__ __launch_bounds__(32)
void ds2x_wmma_kernel(const float* __restrict__ src, float* __restrict__ dst,
                      int Hi, int Wi, int tilesX, int tilesY) {
    __shared__ float sIn[34][36];   // staged input window (reflect-padded)
    __shared__ float sHh[36][16];   // horizontal pass result, K padded to 36

    const int Ho = Hi >> 1, Wo = Wi >> 1;
    int bid = blockIdx.x;
    const int tx = bid % tilesX; bid /= tilesX;
    const int ty = bid % tilesY; bid /= tilesY;
    const int pl = bid;                       // plane = n*C + c
    const int lane = threadIdx.x;

    const float* sp = src + (size_t)pl * Hi * Wi;
    float*       dp = dst + (size_t)pl * Ho * Wo;

    const int yo0 = ty << 4, xo0 = tx << 4;
    const int yIn0 = 2 * yo0 - 1, xIn0 = 2 * xo0 - 1;

    // --- Stage 34x34 input window: async global->LDS (ASYNCcnt tracked) ---
    for (int e = lane; e < 34 * 34; e += 32) {
        const int r = e / 34, q = e - r * 34;
        const int gy = refl(yIn0 + r, Hi);
        const int gx = refl(xIn0 + q, Wi);
        const float* ga = sp + (size_t)gy * Wi + gx;
        // LDS aperture: low 32 bits of the flat pointer == LDS byte offset.
        unsigned lds = (unsigned)(unsigned long long)&sIn[r][q];
        asm volatile("global_load_async_to_lds_b32 %0, %1, off"
                     :: "v"(lds), "v"(ga) : "memory");
    }
    asm volatile("s_wait_asynccnt 0" ::: "memory");
    __syncthreads();

    // --- Horizontal blur pass (VALU, LDS -> LDS) ---
    for (int e = lane; e < 34 * 16; e += 32) {
        const int r = e >> 4, xo = e & 15;
        const int cb = 2 * xo;  // input col = 2*(xo0+xo)-1+dx  ->  local 2*xo+dx
        float h = 0.125f * (sIn[r][cb] + sIn[r][cb + 3])
                + 0.375f * (sIn[r][cb + 1] + sIn[r][cb + 2]);
        sHh[r][xo] = h;
    }
    // zero the K-padding rows 34,35
    sHh[34 + (lane >> 4)][lane & 15] = 0.0f;
    __syncthreads();

    // --- Vertical pass as banded matmul: D[16x16] = A[16x36] x Hh[36x16] ---
    // A[m, kg] = band_coef(kg - 2m)  (tap rows of output m are kg = 2m..2m+3).
    // f32 A 16x4 layout: lanes 0-15 hold K=0,1 in a.x/a.y; lanes 16-31 K=2,3.
    // f32 B 4x16 layout mirrors it: b.x/b.y are rows klBase, klBase+1 at col N.
    const int m      = lane & 15;          // A row / B column / D column
    const int klBase = (lane >> 4) << 1;   // 0 or 2
    v8f acc = {};
    #pragma unroll
    for (int kk = 0; kk < 9; ++kk) {
        const int k0 = 4 * kk + klBase;
        v2f a, b;
        a.x = band_coef(k0 - 2 * m);
        a.y = band_coef(k0 + 1 - 2 * m);
        b.x = sHh[k0][m];
        b.y = sHh[k0 + 1][m];
        acc = __builtin_amdgcn_wmma_f32_16x16x4_f32(
            /*neg_a=*/false, a, /*neg_b=*/false, b,
            /*c_mod=*/(short)0, acc, /*reuse_a=*/false, /*reuse_b=*/false);
    }

    // D layout: VGPR j -> M=j (lanes 0-15) / M=j+8 (lanes 16-31), N = lane&15.
    const int rowHalf = (lane >> 4) * 8;
    #pragma unroll
    for (int j = 0; j < 8; ++j) {
        const int row = j + rowHalf;
        dp[(size_t)(yo0 + row) * Wo + (xo0 + m)] = acc[j];
    }
}

// Per-pixel sample descriptor: warped full-res coord + mip level.
__global__ __launch_bounds__(256)
void params_kernel(const float* __restrict__ grid, float4* __restrict__ prm,
                   int N, int H, int W) {
    const int idx = blockIdx.x * 256 + threadIdx.x;
    if (idx >= N * H * W) return;
    const int x = idx % W;
    const int y = (idx / W) % H;
    const int n = idx / (W * H);

    const float* gb = grid + ((size_t)n * H * W) * 2;
    auto gx_at = [&](int yy, int xx) { return gb[((size_t)yy * W + xx) * 2 + 0]; };
    auto gy_at = [&](int yy, int xx) { return gb[((size_t)yy * W + xx) * 2 + 1]; };

    const float gx = gx_at(y, x), gy = gy_at(y, x);
    const float sx = 0.5f * (W - 1.0f), sy = 0.5f * (H - 1.0f);

    const int xl = max(x - 1, 0), xr = min(x + 1, W - 1);  // edge pad
    const int yu = max(y - 1, 0), yd = min(y + 1, H - 1);

    float maxd2 = 1.0f;
    {
        float dx, dy, d2;
        dx = (gx_at(y, xl) - gx) * sx; dy = (gy_at(y, xl) - gy) * sy;
        d2 = dx * dx + dy * dy; maxd2 = fmaxf(maxd2, d2);
        dx = (gx_at(y, xr) - gx) * sx; dy = (gy_at(y, xr) - gy) * sy;
        d2 = dx * dx + dy * dy; maxd2 = fmaxf(maxd2, d2);
        dx = (gx_at(yu, x) - gx) * sx; dy = (gy_at(yu, x) - gy) * sy;
        d2 = dx * dx + dy * dy; maxd2 = fmaxf(maxd2, d2);
        dx = (gx_at(yd, x) - gx) * sx; dy = (gy_at(yd, x) - gy) * sy;
        d2 = dx * dx + dy * dy; maxd2 = fmaxf(maxd2, d2);
    }
    const float lvl = fminf(fmaxf(0.5f * log2f(maxd2), 0.0f), 2.5f);

    const float ix = fminf(fmaxf(((gx + 1.0f) * W - 1.0f) * 0.5f, 0.0f), W - 1.0f);
    const float iy = fminf(fmaxf(((gy + 1.0f) * H - 1.0f) * 0.5f, 0.0f), H - 1.0f);

    float4 p; p.x = ix; p.y = iy; p.z = lvl; p.w = 0.0f;
    prm[idx] = p;
}

struct Taps {
    const float* p00; const float* p01; const float* p10; const float* p11;
    float w00, w01, w10, w11;
    size_t cs;   // channel stride (plane elements)
};

__device__ __forceinline__ Taps make_taps(const float* base, int lv, int n,
                                          int C, int H, int W,
                                          float ix, float iy) {
    const int Hl = H >> lv, Wl = W >> lv;
    const float scale = 1.0f / (float)(1 << lv);
    // half-pixel-center mapping of full-res coord into the coarse plane
    float cx = fminf(fmaxf((ix + 0.5f) * scale - 0.5f, 0.0f), (float)(Wl - 1));
    float cy = fminf(fmaxf((iy + 0.5f) * scale - 0.5f, 0.0f), (float)(Hl - 1));
    const int x0 = min((int)cx, Wl - 1);
    const int y0 = min((int)cy, Hl - 1);
    const float fx = cx - (float)x0, fy = cy - (float)y0;
    const int x1 = min(x0 + 1, Wl - 1), y1 = min(y0 + 1, Hl - 1);

    Taps t;
    t.cs = (size_t)Hl * Wl;
    const float* b = base + (size_t)n * C * t.cs;
    t.p00 = b + (size_t)y0 * Wl + x0;
    t.p01 = b + (size_t)y0 * Wl + x1;
    t.p10 = b + (size_t)y1 * Wl + x0;
    t.p11 = b + (size_t)y1 * Wl + x1;
    t.w00 = (1.0f - fx) * (1.0f - fy);
    t.w01 = fx * (1.0f - fy);
    t.w10 = (1.0f - fx) * fy;
    t.w11 = fx * fy;
    return t;
}

// One block = one (n, y) row; thread x owns pixel (y,x) across all channels.
__global__ __launch_bounds__(256)
void warp_kernel(const float* __restrict__ lvl0, const float* __restrict__ lvl1,
                 const float* __restrict__ lvl2, const float* __restrict__ lvl3,
                 const float4* __restrict__ prm, float* __restrict__ out,
                 int N, int C, int H, int W) {
    const int x = threadIdx.x;
    const int y = blockIdx.x % H;
    const int n = blockIdx.x / H;

    const float4 p = prm[((size_t)n * H + y) * W + x];
    const float L  = p.z;
    const float lf = floorf(L);
    const int l0 = (int)lf;
    const int l1 = (int)ceilf(L);
    const float wb = L - lf;

    const float* b0 = (l0 == 0) ? lvl0 : (l0 == 1) ? lvl1 : (l0 == 2) ? lvl2 : lvl3;
    const float* b1 = (l1 == 0) ? lvl0 : (l1 == 1) ? lvl1 : (l1 == 2) ? lvl2 : lvl3;

    const Taps A = make_taps(b0, l0, n, C, H, W, p.x, p.y);
    const Taps B = make_taps(b1, l1, n, C, H, W, p.x, p.y);

    float* o = out + (((size_t)n * C) * H + y) * (size_t)W + x;
    const size_t ostride = (size_t)H * W;

    for (int c = 0; c < C; ++c) {
        const size_t oa = (size_t)c * A.cs;
        const size_t ob = (size_t)c * B.cs;
        float v0 = A.w00 * A.p00[oa];
        v0 = fmaf(A.w01, A.p01[oa], v0);
        v0 = fmaf(A.w10, A.p10[oa], v0);
        v0 = fmaf(A.w11, A.p11[oa], v0);
        float v1 = B.w00 * B.p00[ob];
        v1 = fmaf(B.w01, B.p01[ob], v1);
        v1 = fmaf(B.w10, B.p10[ob], v1);
        v1 = fmaf(B.w11, B.p11[ob], v1);
        o[(size_t)c * ostride] = fmaf(wb, v1 - v0, v0);
        // Hide the plane-strided walk to the next channel (global_prefetch_b8).
        __builtin_prefetch(A.p00 + oa + A.cs, 0, 0);
        __builtin_prefetch(B.p00 + ob + B.cs, 0, 0);
    }
}

extern "C" void kernel_launch(void* const* d_in, const int* in_sizes, int n_in,
                              void* d_out, int out_size, void* d_ws, size_t ws_size,
                              hipStream_t stream) {
    (void)in_sizes; (void)n_in; (void)out_size; (void)ws_size;
    const int N = 4, C = 96, H = 256, W = 256;

    const float* inputs = (const float*)d_in[0];
    const float* grid   = (const float*)d_in[1];
    float* out = (float*)d_out;
    float* ws  = (float*)d_ws;

    const size_t sz1 = (size_t)N * C * (H / 2) * (W / 2);
    const size_t sz2 = (size_t)N * C * (H / 4) * (W / 4);
    const size_t sz3 = (size_t)N * C * (H / 8) * (W / 8);
    float*  pyr1 = ws;
    float*  pyr2 = pyr1 + sz1;
    float*  pyr3 = pyr2 + sz2;
    float4* prm  = (float4*)(pyr3 + sz3);

    const int planes = N * C;
    // Pyramid levels 1..3 (each: one wave per 16x16 output tile).
    ds2x_wmma_kernel<<<planes * 8 * 8, 32, 0, stream>>>(inputs, pyr1, H,     W,     8, 8);
    ds2x_wmma_kernel<<<planes * 4 * 4, 32, 0, stream>>>(pyr1,   pyr2, H / 2, W / 2, 4, 4);
    ds2x_wmma_kernel<<<planes * 2 * 2, 32, 0, stream>>>(pyr2,   pyr3, H / 4, W / 4, 2, 2);

    params_kernel<<<(N * H * W) / 256, 256, 0, stream>>>(grid, prm, N, H, W);

    warp_kernel<<<N * H, 256, 0, stream>>>(inputs, pyr1, pyr2, pyr3, prm, out,
                                           N, C, H, W);
}